// AutoregressiveDense_37160057045259
// MI455X (gfx1250) — compile-verified
//
#include <hip/hip_runtime.h>
#include <hip/hip_bf16.h>

typedef __attribute__((ext_vector_type(2))) float v2f;
typedef __attribute__((ext_vector_type(8))) float v8f;
typedef int v4i_ __attribute__((vector_size(16)));   // payload type for async b128

#define B_SZ   8192
#define D_SZ   1024
#define L_SZ   64
#define O_SZ   64
#define STRIDE 16
#define KC     64          // K-chunk staged in LDS per iteration
#define BM     64          // batch rows per block (4 M-tiles of 16)
#define XP     (KC + 4)    // xs row stride: A b64 reads hit all 64 banks once
#define WP     (O_SZ + 8)  // ws row stride: B b32 reads conflict-free (16*half+idx)

// Async global->LDS path (ASYNCcnt) if this toolchain declares the builtins;
// otherwise fall back to plain global->VGPR->ds_store staging.
#if defined(__HIP_DEVICE_COMPILE__) && \
    __has_builtin(__builtin_amdgcn_global_load_async_to_lds_b128)
#define USE_ASYNC 1
#define ASYNC_CP_B128(gsrc, ldst) \
    __builtin_amdgcn_global_load_async_to_lds_b128((v4i_*)(gsrc), (v4i_*)(ldst), 0, 0)
#if __has_builtin(__builtin_amdgcn_s_wait_asynccnt)
#define WAIT_ASYNC(n) __builtin_amdgcn_s_wait_asynccnt((unsigned short)(n))
#else
#define WAIT_ASYNC(n) asm volatile("s_wait_asynccnt %0" ::"i"(n) : "memory")
#endif
#else
#define USE_ASYNC 0
#define WAIT_ASYNC(n) ((void)0)
#endif

// Stage one K-chunk: x tile 64x64 floats + W tile 64x64 floats.
// 256 threads, 4 float4 each per tile -> 8 async b128 instructions per wave.
// Fully uniform (no branches) so ASYNCcnt bookkeeping is exact: rows beyond the
// causal boundary are staged but never read (compute loop is K-trimmed), and
// kc+63 <= 1023 keeps every global read in-bounds.
__device__ __forceinline__ void stage_chunk(
    const float* __restrict__ x, const float* __restrict__ Wl,
    int rowBase, int kc, int tid, float* xsBuf, float* wsBuf)
{
#pragma unroll
    for (int r = 0; r < 4; ++r) {
        const int t   = tid + r * 256;
        const int row = t >> 4;          // 0..63
        const int col = (t & 15) * 4;    // 0..60
        const float* g = x + (size_t)(rowBase + row) * D_SZ + kc + col;
        float* l = xsBuf + row * XP + col;
#if USE_ASYNC
        ASYNC_CP_B128(g, l);
#else
        *(float4*)l = *(const float4*)g;
#endif
    }
#pragma unroll
    for (int r = 0; r < 4; ++r) {
        const int t   = tid + r * 256;
        const int row = t >> 4;          // k within chunk, 0..63
        const int col = (t & 15) * 4;
        const float* g = Wl + (size_t)(kc + row) * O_SZ + col;
        float* l = wsBuf + row * WP + col;
#if USE_ASYNC
        ASYNC_CP_B128(g, l);
#else
        *(float4*)l = *(const float4*)g;
#endif
    }
}

// out[b, l, o] = sum_{d < 16*l} x[b,d] * W[l,d,o] + bias[l,o]
// Block: 64 batch rows x all 64 out cols for one layer; 8 waves, each wave owns
// two 16x16 M-tiles (shared B operand) x one 16-col N-tile.
__global__ __launch_bounds__(256) void ard_wmma_f32_kernel(
    const float* __restrict__ x,     // [B, D]
    const float* __restrict__ W,     // [L, D, O]
    const float* __restrict__ bias,  // [L, O]
    float* __restrict__ out)         // [B, L, O]
{
    __shared__ float xs[2][BM][XP];  // double-buffered x chunk
    __shared__ float ws[2][KC][WP];  // double-buffered W chunk

    const int layer   = blockIdx.y;
    const int rowBase = blockIdx.x * BM;
    const int vis     = layer * STRIDE;   // causal visibility bound

    const int tid  = threadIdx.x;
    const int lane = tid & 31;
    const int wave = tid >> 5;
    const int nt   = wave & 3;    // N tile 0..3
    const int mtp  = wave >> 2;   // M tile pair 0..1 (rows mtp*32 .. mtp*32+31)
    const int idx  = lane & 15;   // M (A) / N (B,C,D)
    const int half = lane >> 4;   // K sub-pair select per ISA 16x4/4x16 f32 layout

    const float* Wl = W + (size_t)layer * D_SZ * O_SZ;

    v8f c0 = {};
    v8f c1 = {};

    const int nch = (vis + KC - 1) / KC;
    if (nch > 0) {
        stage_chunk(x, Wl, rowBase, 0, tid, &xs[0][0][0], &ws[0][0][0]);
    }

    for (int ch = 0; ch < nch; ++ch) {
        const int buf = ch & 1;

        // Prefetch next chunk into the other buffer, then wait for this one.
        if (ch + 1 < nch) {
            stage_chunk(x, Wl, rowBase, (ch + 1) * KC, tid,
                        &xs[buf ^ 1][0][0], &ws[buf ^ 1][0][0]);
            WAIT_ASYNC(8);   // 8 outstanding = next stage only -> this stage done
        } else {
            WAIT_ASYNC(0);
        }
        __syncthreads();

        int klen = vis - ch * KC;         // multiple of 16
        if (klen > KC) klen = KC;

        // A 16x4 f32: lane holds A[idx, 2*half + {0,1}]   (one ds b64)
        // B 4x16 f32: lane holds B[2*half + {0,1}, idx]   (two ds b32)
        const float* xrow0 = &xs[buf][mtp * 32 + idx][2 * half];
        const float* xrow1 = xrow0 + 16 * XP;
        const float* wcol  = &ws[buf][2 * half][nt * 16 + idx];

        for (int k16 = 0; k16 < klen; k16 += 16) {
#pragma unroll
            for (int kk = 0; kk < 16; kk += 4) {
                const int k = k16 + kk;
                v2f a0 = *(const v2f*)(xrow0 + k);
                v2f a1 = *(const v2f*)(xrow1 + k);
                v2f b;
                b.x = wcol[k * WP];
                b.y = wcol[(k + 1) * WP];
                c0 = __builtin_amdgcn_wmma_f32_16x16x4_f32(
                         false, a0, false, b, (short)0, c0, false, false);
                c1 = __builtin_amdgcn_wmma_f32_16x16x4_f32(
                         false, a1, false, b, (short)0, c1, false, false);
            }
        }
        __syncthreads();
    }

    // Epilogue: add bias, store D (lane l, VGPR r -> M = r + 8*half, N = idx)
    const int   n  = nt * 16 + idx;
    const float bv = bias[(size_t)layer * O_SZ + n];
    float* op0 = out + (size_t)(rowBase + mtp * 32 + 8 * half) * (L_SZ * O_SZ)
                     + (size_t)layer * O_SZ + n;
    float* op1 = op0 + (size_t)16 * (L_SZ * O_SZ);
#pragma unroll
    for (int r = 0; r < 8; ++r) {
        op0[(size_t)r * (L_SZ * O_SZ)] = c0[r] + bv;
        op1[(size_t)r * (L_SZ * O_SZ)] = c1[r] + bv;
    }
}

extern "C" void kernel_launch(void* const* d_in, const int* in_sizes, int n_in,
                              void* d_out, int out_size, void* d_ws, size_t ws_size,
                              hipStream_t stream) {
    (void)in_sizes; (void)n_in; (void)d_ws; (void)ws_size; (void)out_size;
    const float* x    = (const float*)d_in[0];  // [8192, 1024]
    const float* W    = (const float*)d_in[1];  // [64, 1024, 64]
    const float* bias = (const float*)d_in[2];  // [64, 64]
    float* out        = (float*)d_out;          // [8192, 64, 64]

    dim3 grid(B_SZ / BM, L_SZ, 1);
    ard_wmma_f32_kernel<<<grid, 256, 0, stream>>>(x, W, bias, out);
}